// MultiRelGraphLayer_23862838297344
// MI455X (gfx1250) — compile-verified
//
#include <hip/hip_runtime.h>
#include <hip/hip_bf16.h>

// MI455X / gfx1250: wave32, WMMA f32 16x16x4 (exact fp32 semantics vs reference).
// Pipeline:
//   0) memsetAsync accumulator+degree (in d_ws)
//   1) edge_msg_kernel: 64-edge x 128-out tile per block, 8 waves x 4 accumulators,
//      K=256 WMMA chain; A-tile staged to LDS via GLOBAL_LOAD_ASYNC_TO_LDS_B128
//      (ASYNCcnt-tracked DMA, bypasses VGPRs); degree counting fused in;
//      scatter-accumulate into L2-resident [N,128] accumulator.
//      b1 deferred to pass 2 (sum(msg)+deg*b1 identity).
//   2) node_out_kernel: node_msg = acc/max(deg,1) + b1*(deg>0) staged in LDS,
//      32-node x 128-out tile, K=128 WMMA, residual + b2 + RReLU(11/48),
//      non-temporal store.

typedef __attribute__((ext_vector_type(2))) float v2f;
typedef __attribute__((ext_vector_type(4))) float v4f;
typedef __attribute__((ext_vector_type(8))) float v8f;

// Exact parameter type expected by the async-LDS builtin (from round-3 diag):
// int __attribute__((vector_size(16))) in AS1 (global) / AS3 (LDS).
typedef int vint4 __attribute__((vector_size(4 * sizeof(int))));
typedef __attribute__((address_space(1))) vint4 as1_vint4;
typedef __attribute__((address_space(3))) vint4 as3_vint4;

#define DD 128          // feature dim
#define HH 128          // hidden dim
#define K1 256          // 2*D
#define A_STRIDE 260    // 16B-aligned; stride%64==4 -> conflict-free frag reads
#define X_STRIDE 132    // same property
#define M_EDGE 64       // edges per block
#define M_NODE 32       // nodes per block

#define RRELU_SLOPE 0.2291666666666667f  // 11/48

#if defined(__has_builtin)
#if __has_builtin(__builtin_amdgcn_global_load_async_to_lds_b128) && \
    __has_builtin(__builtin_amdgcn_s_wait_asynccnt)
#define USE_ASYNC_LDS 1
#endif
#endif

#ifdef USE_ASYNC_LDS
__device__ __forceinline__ void async_copy16(const float* g, float* l) {
  __builtin_amdgcn_global_load_async_to_lds_b128(
      (as1_vint4*)g, (as3_vint4*)l, /*imm_offset=*/0, /*cpol=*/0);
}
#endif

// Block: 64 edges x 128 outputs. 8 waves; each wave: 4 stacked 16x16 tiles sharing
// one B fragment per K-step -> 4 WMMAs per 2 global dword loads.
__global__ __launch_bounds__(256) void edge_msg_kernel(
    const float* __restrict__ node_feats,
    const float* __restrict__ edge_feats,
    const int* __restrict__ src,
    const int* __restrict__ dst,
    const float* __restrict__ W1,   // [256,128] row-major
    float* __restrict__ msg_sum,    // [N,128] accumulator (pre-zeroed)
    float* __restrict__ deg,        // [N] degree accumulator (pre-zeroed)
    int E) {
  __shared__ float Atile[M_EDGE * A_STRIDE];
  __shared__ int   ssrc[M_EDGE];
  __shared__ int   sdst[M_EDGE];    // dst row base offsets (dst*128)

  const int tid = threadIdx.x;
  const int e0  = blockIdx.x * M_EDGE;
  const bool full = (e0 + M_EDGE <= E);

  if (tid < M_EDGE) {
    const int e = e0 + tid;
    if (e < E) {
      const int d = dst[e];
      ssrc[tid] = src[e];
      sdst[tid] = d * HH;
      atomicAdd(deg + d, 1.0f);       // fused degree count
    } else {
      ssrc[tid] = 0;
      sdst[tid] = 0;
    }
  }
  __syncthreads();

  // Stage A: row e -> [node_feats[src[e]] (128) | edge_feats[e] (128)]
  // 64 rows x 64 float4 chunks = 4096 chunks / 256 threads = 16 iters.
  for (int idx = tid; idx < M_EDGE * 64; idx += 256) {
    const int row = idx >> 6;
    const int col = (idx & 63) << 2;
    const int e = e0 + row;
    float* ldst = Atile + row * A_STRIDE + col;
    if (e < E) {
      const float* g = (col < DD)
          ? node_feats + (long)ssrc[row] * DD + col          // L2-resident gather
          : edge_feats + (long)e * DD + (col - DD);          // streamed once
#ifdef USE_ASYNC_LDS
      async_copy16(g, ldst);                                  // DMA, bypass VGPRs
#else
      v4f v = (col < DD) ? *(const v4f*)g
                         : __builtin_nontemporal_load((const v4f*)g);
      *(v4f*)ldst = v;
#endif
    } else {
      *(v4f*)ldst = (v4f){0.f, 0.f, 0.f, 0.f};
    }
  }
#ifdef USE_ASYNC_LDS
  __builtin_amdgcn_s_wait_asynccnt(0);
#endif
  __syncthreads();

  const int wave = tid >> 5;
  const int lane = tid & 31;
  const int half = lane >> 4;   // 0: K pair {0,1}, 1: K pair {2,3}
  const int lm   = lane & 15;
  const int n    = wave * 16 + lm;

  v8f c[4] = {};
#pragma unroll 4
  for (int k0 = 0; k0 < K1; k0 += 4) {
    const int kk = k0 + 2 * half;
    v2f b;
    b.x = W1[kk * HH + n];
    b.y = W1[(kk + 1) * HH + n];
    const float* ap = Atile + lm * A_STRIDE + kk;
    v2f a0 = *(const v2f*)(ap);
    v2f a1 = *(const v2f*)(ap + 16 * A_STRIDE);
    v2f a2 = *(const v2f*)(ap + 32 * A_STRIDE);
    v2f a3 = *(const v2f*)(ap + 48 * A_STRIDE);
    c[0] = __builtin_amdgcn_wmma_f32_16x16x4_f32(false, a0, false, b, (short)0, c[0], false, false);
    c[1] = __builtin_amdgcn_wmma_f32_16x16x4_f32(false, a1, false, b, (short)0, c[1], false, false);
    c[2] = __builtin_amdgcn_wmma_f32_16x16x4_f32(false, a2, false, b, (short)0, c[2], false, false);
    c[3] = __builtin_amdgcn_wmma_f32_16x16x4_f32(false, a3, false, b, (short)0, c[3], false, false);
  }

  // C layout: VGPR r -> M = r + 8*half (within each 16-row sub-tile), N = n.
  if (full) {
#pragma unroll
    for (int i = 0; i < 4; ++i) {
#pragma unroll
      for (int r = 0; r < 8; ++r) {
        const int m = i * 16 + r + 8 * half;
        atomicAdd(msg_sum + sdst[m] + n, c[i][r]);
      }
    }
  } else {
#pragma unroll
    for (int i = 0; i < 4; ++i) {
#pragma unroll
      for (int r = 0; r < 8; ++r) {
        const int m = i * 16 + r + 8 * half;
        if (e0 + m < E) atomicAdd(msg_sum + sdst[m] + n, c[i][r]);
      }
    }
  }
}

// Block: 32 nodes x 128 outputs. node_msg staged (scaled, +b1 gated by deg>0).
__global__ __launch_bounds__(256) void node_out_kernel(
    const float* __restrict__ msg_sum,
    const float* __restrict__ deg,
    const float* __restrict__ b1,   // [128]  (deferred edge-GEMM bias)
    const float* __restrict__ W2,   // [128,128] row-major
    const float* __restrict__ b2,   // [128]
    float* __restrict__ out,        // [N,128]
    int N) {
  __shared__ float X[M_NODE * X_STRIDE];

  const int tid = threadIdx.x;
  const int node0 = blockIdx.x * M_NODE;
  const bool full = (node0 + M_NODE <= N);

  // 32 rows x 32 float4 chunks = 1024 chunks / 256 threads = 4 iters.
  for (int idx = tid; idx < M_NODE * 32; idx += 256) {
    const int row = idx >> 5;
    const int col = (idx & 31) << 2;
    const int node = node0 + row;
    v4f v = {0.f, 0.f, 0.f, 0.f};
    if (node < N) {
      const float dg   = deg[node];
      const float inv  = 1.0f / fmaxf(dg, 1.0f);
      const float gate = (dg > 0.f) ? 1.0f : 0.0f;   // b1 only for non-isolated
      v4f s  = *(const v4f*)(msg_sum + (long)node * HH + col);
      v4f bb = *(const v4f*)(b1 + col);
      v = s * inv + bb * gate;
    }
    *(v4f*)(X + row * X_STRIDE + col) = v;
  }
  __syncthreads();

  const int wave = tid >> 5;
  const int lane = tid & 31;
  const int half = lane >> 4;
  const int lm   = lane & 15;
  const int n    = wave * 16 + lm;

  v8f c[2] = {};
#pragma unroll 4
  for (int k0 = 0; k0 < DD; k0 += 4) {
    const int kk = k0 + 2 * half;
    v2f b;
    b.x = W2[kk * HH + n];
    b.y = W2[(kk + 1) * HH + n];
    const float* xp = X + lm * X_STRIDE + kk;
    v2f a0 = *(const v2f*)(xp);
    v2f a1 = *(const v2f*)(xp + 16 * X_STRIDE);
    c[0] = __builtin_amdgcn_wmma_f32_16x16x4_f32(false, a0, false, b, (short)0, c[0], false, false);
    c[1] = __builtin_amdgcn_wmma_f32_16x16x4_f32(false, a1, false, b, (short)0, c[1], false, false);
  }

  const float bias = b2[n];
#pragma unroll
  for (int i = 0; i < 2; ++i) {
#pragma unroll
    for (int r = 0; r < 8; ++r) {
      const int m = i * 16 + r + 8 * half;
      const int node = node0 + m;
      if (full || node < N) {
        float val = X[m * X_STRIDE + n] + c[i][r] + bias;  // residual + W2 + b2
        val = (val >= 0.f) ? val : RRELU_SLOPE * val;      // RReLU eval
        __builtin_nontemporal_store(val, out + (long)node * HH + n);
      }
    }
  }
}

extern "C" void kernel_launch(void* const* d_in, const int* in_sizes, int n_in,
                              void* d_out, int out_size, void* d_ws, size_t ws_size,
                              hipStream_t stream) {
  const float* node_feats = (const float*)d_in[0];
  const float* edge_feats = (const float*)d_in[1];
  const int*   src        = (const int*)d_in[2];
  const int*   dst        = (const int*)d_in[3];
  const float* W1         = (const float*)d_in[4];
  const float* b1         = (const float*)d_in[5];
  const float* W2         = (const float*)d_in[6];
  const float* b2         = (const float*)d_in[7];

  const int N = in_sizes[0] / DD;
  const int E = in_sizes[2];

  float* msg_sum = (float*)d_ws;                    // [N,128]
  float* deg     = msg_sum + (size_t)N * HH;        // [N]

  (void)hipMemsetAsync(d_ws, 0, ((size_t)N * HH + (size_t)N) * sizeof(float), stream);

  edge_msg_kernel<<<(E + M_EDGE - 1) / M_EDGE, 256, 0, stream>>>(
      node_feats, edge_feats, src, dst, W1, msg_sum, deg, E);
  node_out_kernel<<<(N + M_NODE - 1) / M_NODE, 256, 0, stream>>>(
      msg_sum, deg, b1, W2, b2, (float*)d_out, N);
}